// PointCNN_36541581754659
// MI455X (gfx1250) — compile-verified
//
#include <hip/hip_runtime.h>
#include <cmath>

typedef float v2f __attribute__((ext_vector_type(2)));
typedef float v8f __attribute__((ext_vector_type(8)));

#define BN   16
#define NPTS 2048

__device__ __forceinline__ float eluf(float x) { return x > 0.f ? x : expm1f(x); }

// ---------------------------------------------------------------------------
// Transpose x (B,3,N) -> pts (B,N,3)
// ---------------------------------------------------------------------------
__global__ void transpose_pts(const float* __restrict__ x, float* __restrict__ pts, int total) {
    int gid = blockIdx.x * blockDim.x + threadIdx.x;
    if (gid >= total) return;                 // total = B*NPTS
    int n = gid % NPTS, b = gid / NPTS;
    const float* xb = x + (size_t)b * 3 * NPTS;
    pts[(size_t)gid * 3 + 0] = xb[0 * NPTS + n];
    pts[(size_t)gid * 3 + 1] = xb[1 * NPTS + n];
    pts[(size_t)gid * 3 + 2] = xb[2 * NPTS + n];
}

// ---------------------------------------------------------------------------
// KNN: per query, insertion-select kd = K*dil smallest distances
// (ascending d, ties keep smaller index first — matches jax top_k(-d)),
// then emit every dil-th index.
// ---------------------------------------------------------------------------
__global__ void knn_kernel(const float* __restrict__ pts, int* __restrict__ idxOut,
                           int Mq, int P, int K, int dil, int total) {
    int gid = blockIdx.x * blockDim.x + threadIdx.x;
    if (gid >= total) return;                 // total = B*Mq
    int m = gid % Mq, b = gid / Mq;
    const float* pb = pts + (size_t)b * NPTS * 3;
    float qx = pb[m * 3 + 0], qy = pb[m * 3 + 1], qz = pb[m * 3 + 2];
    float bd[48]; int bi[48];
    int kd = K * dil, cnt = 0;
    for (int n = 0; n < P; ++n) {
        float dx = pb[n * 3 + 0] - qx;
        float dy = pb[n * 3 + 1] - qy;
        float dz = pb[n * 3 + 2] - qz;
        float d = dx * dx + dy * dy + dz * dz;
        if (cnt == kd && d >= bd[kd - 1]) continue;
        int j = (cnt < kd) ? cnt : (kd - 1);
        while (j > 0 && bd[j - 1] > d) { bd[j] = bd[j - 1]; bi[j] = bi[j - 1]; --j; }
        bd[j] = d; bi[j] = n;
        if (cnt < kd) ++cnt;
    }
    for (int k = 0; k < K; ++k) idxOut[(size_t)gid * K + k] = bi[k * dil];
}

// ---------------------------------------------------------------------------
// r[b,m,k,:] = pts[b, idx[b,m,k], :] - pts[b, m, :]
// stored as (B*M, K*3) == (B*M*K, 3)
// ---------------------------------------------------------------------------
__global__ void build_r(const float* __restrict__ pts, const int* __restrict__ idx,
                        float* __restrict__ r, int Mq, int K, int total) {
    int gid = blockIdx.x * blockDim.x + threadIdx.x;
    if (gid >= total) return;                 // total = B*Mq*K
    int bm = gid / K;
    int m = bm % Mq, b = bm / Mq;
    int n = idx[gid];
    const float* pb = pts + (size_t)b * NPTS * 3;
    r[(size_t)gid * 3 + 0] = pb[n * 3 + 0] - pb[m * 3 + 0];
    r[(size_t)gid * 3 + 1] = pb[n * 3 + 1] - pb[m * 3 + 1];
    r[(size_t)gid * 3 + 2] = pb[n * 3 + 2] - pb[m * 3 + 2];
}

// ---------------------------------------------------------------------------
// F[row, mid + c] = featsPrev[b, idx[row], c]   (row = (b*M+m)*K + k)
// ---------------------------------------------------------------------------
__global__ void gather_feats(const float* __restrict__ feats, const int* __restrict__ idx,
                             float* __restrict__ F, int Mq, int K, int mid, int cf,
                             int Ctot, int Pprev, int total) {
    int gid = blockIdx.x * blockDim.x + threadIdx.x;
    if (gid >= total) return;                 // total = B*Mq*K*cf
    int c = gid % cf;
    int t = gid / cf;                         // (b*Mq + m)*K + k
    int b = (t / K) / Mq;
    int n = idx[t];
    F[(size_t)t * Ctot + mid + c] = feats[((size_t)b * Pprev + n) * cf + c];
}

// ---------------------------------------------------------------------------
// Naive GEMM (small N): C[row, col] = A(row-major, lda=Kin) x B(Kin x Ncols)
// ---------------------------------------------------------------------------
__global__ void naive_gemm_f32(const float* __restrict__ A, const float* __restrict__ Bw,
                               float* __restrict__ C, int Mrows, int Ncols, int Kin,
                               int ldC, int applyElu) {
    int gid = blockIdx.x * blockDim.x + threadIdx.x;
    int total = Mrows * Ncols;
    if (gid >= total) return;
    int row = gid / Ncols, col = gid % Ncols;
    const float* Ar = A + (size_t)row * Kin;
    float acc = 0.f;
    for (int k = 0; k < Kin; ++k) acc = fmaf(Ar[k], Bw[(size_t)k * Ncols + col], acc);
    if (applyElu) acc = eluf(acc);
    C[(size_t)row * ldC + col] = acc;
}

// ---------------------------------------------------------------------------
// Repack weight B (Kin x Ncols, row-major) into pair-interleaved layout:
//   Bp[p*Ncols + col] = { B[2p][col], B[2p+1][col] }   (v2f pairs)
// so a WMMA B-fragment (rows k+2hi, k+2hi+1 at one column) is ONE aligned
// 8-byte load in the GEMM inner loop instead of two strided b32 loads.
// ---------------------------------------------------------------------------
__global__ void pack_pairs(const float* __restrict__ Bw, float* __restrict__ Bp,
                           int Ncols, int total) {
    int gid = blockIdx.x * blockDim.x + threadIdx.x;
    if (gid >= total) return;                 // total = (Kin/2)*Ncols
    int col = gid % Ncols, p = gid / Ncols;
    v2f v;
    v.x = Bw[(size_t)(2 * p) * Ncols + col];
    v.y = Bw[(size_t)(2 * p + 1) * Ncols + col];
    *(v2f*)(Bp + (size_t)gid * 2) = v;
}

// ---------------------------------------------------------------------------
// fp32 WMMA GEMM on pair-packed B. Each wave owns a 16x64 tile of C: one A
// fragment per k-step feeds 4 back-to-back V_WMMA_F32_16X16X4_F32
// (4 accumulators). All fragment loads are b64; the k-loop is double-buffered
// so the next step's 5 loads are in flight while the current 4 WMMAs execute.
// Block = 4 waves covering 4 row tiles (64 rows). Requires Mrows%64==0,
// Kin%4==0. Out-of-range columns are clamped on load, guarded on store, so
// there is no divergence before the WMMAs (EXEC all 1s).
// ---------------------------------------------------------------------------
__global__ __launch_bounds__(128) void wmma_gemm_f32(
        const float* __restrict__ A, const float* __restrict__ Bpack,
        float* __restrict__ C, int Mrows, int Ncols, int Kin, int ldC, int applyElu) {
    const int lane = threadIdx.x;             // 0..31 (one wave per threadIdx.y)
    const int hi   = lane >> 4;               // 0 or 1
    const int lm   = lane & 15;
    const int rowBase = (blockIdx.x * 4 + threadIdx.y) * 16;
    const int colBase = blockIdx.y * 64;

    v8f acc[4];
#pragma unroll
    for (int t = 0; t < 4; ++t) acc[t] = (v8f){0.f,0.f,0.f,0.f,0.f,0.f,0.f,0.f};

    const float* Arow = A + (size_t)(rowBase + lm) * Kin + 2 * hi;
    const v2f* Bp2 = (const v2f*)Bpack;
    const v2f* Bt[4];
    int colT[4];
#pragma unroll
    for (int t = 0; t < 4; ++t) {
        int c = colBase + t * 16 + lm;
        colT[t] = c;
        int cc = c < Ncols ? c : (Ncols - 1);       // clamp, store is guarded
        Bt[t] = Bp2 + (size_t)hi * Ncols + cc;      // pair row (k/2 + hi), col cc
    }

    // Software-pipelined k-loop: fragments for step k+4 are loaded before the
    // WMMAs of step k are issued.
    v2f a0 = *(const v2f*)(Arow);
    v2f b0[4];
#pragma unroll
    for (int t = 0; t < 4; ++t) b0[t] = Bt[t][0];

    for (int k = 0; k < Kin; k += 4) {
        int kn = (k + 4 < Kin) ? (k + 4) : k;       // tail: redundant reload
        v2f a1 = *(const v2f*)(Arow + kn);
        v2f b1[4];
#pragma unroll
        for (int t = 0; t < 4; ++t) b1[t] = Bt[t][(size_t)(kn >> 1) * Ncols];
#pragma unroll
        for (int t = 0; t < 4; ++t)
            acc[t] = __builtin_amdgcn_wmma_f32_16x16x4_f32(false, a0, false, b0[t],
                                                           (short)0, acc[t], false, false);
        a0 = a1;
#pragma unroll
        for (int t = 0; t < 4; ++t) b0[t] = b1[t];
    }
#pragma unroll
    for (int t = 0; t < 4; ++t) {
        if (colT[t] < Ncols) {
#pragma unroll
            for (int i = 0; i < 8; ++i) {
                float v = acc[t][i];
                if (applyElu) v = eluf(v);
                C[(size_t)(rowBase + i + 8 * hi) * ldC + colT[t]] = v;
            }
        }
    }
}

// ---------------------------------------------------------------------------
// Batched einsum Fx[b,m,k,c] = sum_j X[b,m,k,j] * F[b,m,j,c].
// One wave per point (b*M+m). The X (A-side) fragments are invariant across
// all column tiles of F, so they are loaded ONCE into registers and reused
// for every 16-col tile (up to 18 tiles * K/4 WMMAs per X fetch at layer 3).
// K in {8,12,16} (multiple of 4): templated so jstep loops fully unroll.
// ---------------------------------------------------------------------------
template <int K>
__global__ __launch_bounds__(32) void xf_wmma(
        const float* __restrict__ Xm, const float* __restrict__ F,
        float* __restrict__ Fx, int C) {
    const int lane = threadIdx.x;
    const int hi   = lane >> 4;
    const int lm   = lane & 15;
    const int row  = blockIdx.x;              // b*M + m
    const float* Xrow = Xm + (size_t)row * K * K;
    const float* Frow = F + (size_t)row * K * C;
    const bool rowOK = lm < K;                // output row (k index) valid

    constexpr int JS = K / 4;
    v2f a[JS];
#pragma unroll
    for (int js = 0; js < JS; ++js) {
        int jb = js * 4 + 2 * hi;             // jb, jb+1 < K always
        a[js].x = rowOK ? Xrow[lm * K + jb] : 0.f;
        a[js].y = rowOK ? Xrow[lm * K + jb + 1] : 0.f;
    }

    const int nct = (C + 15) >> 4;
    for (int ct = 0; ct < nct; ++ct) {
        int c = ct * 16 + lm;
        int cc = c < C ? c : (C - 1);         // clamp loads, guard stores
        v8f acc = {0.f, 0.f, 0.f, 0.f, 0.f, 0.f, 0.f, 0.f};
#pragma unroll
        for (int js = 0; js < JS; ++js) {
            int jb = js * 4 + 2 * hi;
            v2f b;
            b.x = Frow[(size_t)jb * C + cc];
            b.y = Frow[(size_t)(jb + 1) * C + cc];
            acc = __builtin_amdgcn_wmma_f32_16x16x4_f32(false, a[js], false, b,
                                                        (short)0, acc, false, false);
        }
        if (c < C) {
#pragma unroll
            for (int i = 0; i < 8; ++i) {
                int kk = i + 8 * hi;
                if (kk < K) Fx[(size_t)row * K * C + (size_t)kk * C + c] = acc[i];
            }
        }
    }
}

// ---------------------------------------------------------------------------
// out[b,c] = mean_m( h[b,m,:] @ Wfc[:,c] ) + bfc[c]
// ---------------------------------------------------------------------------
__global__ void final_fc_mean(const float* __restrict__ h, const float* __restrict__ Wfc,
                              const float* __restrict__ bfc, float* __restrict__ out,
                              int total) {
    int gid = blockIdx.x * blockDim.x + threadIdx.x;
    if (gid >= total) return;                 // total = B*40
    int c = gid % 40, b = gid / 40;
    const float* hb = h + (size_t)b * 128 * 128;
    float acc = 0.f;
    for (int m = 0; m < 128; ++m) {
        const float* hr = hb + m * 128;
        float s = 0.f;
        for (int kk = 0; kk < 128; ++kk) s = fmaf(hr[kk], Wfc[kk * 40 + c], s);
        acc += s;
    }
    out[gid] = acc * (1.f / 128.f) + bfc[c];
}

// ---------------------------------------------------------------------------
// Host orchestration
// ---------------------------------------------------------------------------
static inline void launch_wmma(const float* A, const float* Bw, float* Bpack, float* C,
                               int M, int N, int K, int ldC, int elu, hipStream_t s) {
    int ptotal = (K / 2) * N;
    pack_pairs<<<(ptotal + 255) / 256, 256, 0, s>>>(Bw, Bpack, N, ptotal);
    dim3 g((unsigned)(M / 64), (unsigned)((N + 63) / 64));
    dim3 b(32, 4);
    wmma_gemm_f32<<<g, b, 0, s>>>(A, Bpack, C, M, N, K, ldC, elu);
}
static inline void launch_naive(const float* A, const float* Bw, float* C,
                                int M, int N, int K, int ldC, int elu, hipStream_t s) {
    int total = M * N;
    naive_gemm_f32<<<(total + 255) / 256, 256, 0, s>>>(A, Bw, C, M, N, K, ldC, elu);
}
static inline void launch_xf(const float* Xm, const float* F, float* Fx,
                             int rowsBM, int K, int C, hipStream_t s) {
    dim3 g((unsigned)rowsBM);
    if (K == 8)       xf_wmma<8><<<g, 32, 0, s>>>(Xm, F, Fx, C);
    else if (K == 12) xf_wmma<12><<<g, 32, 0, s>>>(Xm, F, Fx, C);
    else              xf_wmma<16><<<g, 32, 0, s>>>(Xm, F, Fx, C);
}

extern "C" void kernel_launch(void* const* d_in, const int* in_sizes, int n_in,
                              void* d_out, int out_size, void* d_ws, size_t ws_size,
                              hipStream_t stream) {
    (void)in_sizes; (void)n_in; (void)out_size; (void)ws_size;
    const float* x = (const float*)d_in[0];
    struct LW { const float *Wd1, *Wd2, *Wx1, *Wx2, *Wx3, *Wc; } lw[4];
    for (int i = 0; i < 4; ++i) {
        lw[i].Wd1 = (const float*)d_in[1 + 6 * i + 0];
        lw[i].Wd2 = (const float*)d_in[1 + 6 * i + 1];
        lw[i].Wx1 = (const float*)d_in[1 + 6 * i + 2];
        lw[i].Wx2 = (const float*)d_in[1 + 6 * i + 3];
        lw[i].Wx3 = (const float*)d_in[1 + 6 * i + 4];
        lw[i].Wc  = (const float*)d_in[1 + 6 * i + 5];
    }
    const float* Wm1 = (const float*)d_in[25];
    const float* Wm2 = (const float*)d_in[26];
    const float* Wfc = (const float*)d_in[27];
    const float* bfc = (const float*)d_in[28];
    float* out = (float*)d_out;

    // Bump allocator over workspace (max sizes across layers)
    size_t off = 0;
    auto alloc = [&](size_t nbytes) -> void* {
        void* p = (char*)d_ws + off;
        off += (nbytes + 255) & ~(size_t)255;
        return p;
    };
    float* pts    = (float*)alloc(sizeof(float) * (size_t)BN * NPTS * 3);
    float* featsA = (float*)alloc(sizeof(float) * 786432);   // max B*M*co
    float* featsB = (float*)alloc(sizeof(float) * 786432);
    int*   idx    = (int*)alloc(sizeof(int) * 131072);       // max B*M*K
    float* r      = (float*)alloc(sizeof(float) * 393216);   // max B*M*K*3
    float* fdt    = (float*)alloc(sizeof(float) * 3145728);  // max B*M*K*mid
    float* Fbuf   = (float*)alloc(sizeof(float) * 9437184);  // max B*M*K*Ctot
    float* t1     = (float*)alloc(sizeof(float) * 1048576);  // max B*M*K*K
    float* t2     = (float*)alloc(sizeof(float) * 1048576);
    float* Xm     = (float*)alloc(sizeof(float) * 1048576);
    float* Fx     = (float*)alloc(sizeof(float) * 9437184);  // max B*M*K*Ctot
    float* wpack  = (float*)alloc(sizeof(float) * 1769472);  // max Kin*N (Wc3)
    float* h1     = t1;  // reused after the layer loop (524288 <= 1048576)
    float* h2     = t2;  // (262144 <= 1048576)

    transpose_pts<<<(BN * NPTS + 255) / 256, 256, 0, stream>>>(x, pts, BN * NPTS);

    static const int CO[4]  = {48, 96, 192, 384};
    static const int MM[4]  = {1024, 384, 128, 128};
    static const int KK[4]  = {8, 12, 16, 16};
    static const int DIL[4] = {1, 2, 2, 3};
    static const int CF[4]  = {0, 48, 96, 192};

    int P = NPTS;                  // points available this layer (prev layer's M)
    const float* featsPrev = nullptr;
    const float* ffinal = nullptr;

    for (int i = 0; i < 4; ++i) {
        const int co = CO[i], M = MM[i], K = KK[i], dil = DIL[i], cf = CF[i];
        const int mid = co / 4, Ctot = mid + cf;
        const int rowsBM = BN * M, rowsBMK = rowsBM * K;
        float* fo = (i & 1) ? featsB : featsA;

        knn_kernel<<<(rowsBM + 127) / 128, 128, 0, stream>>>(pts, idx, M, P, K, dil, rowsBM);
        build_r<<<(rowsBMK + 255) / 256, 256, 0, stream>>>(pts, idx, r, M, K, rowsBMK);

        // fd = elu(elu(r @ Wd1) @ Wd2) -> columns [0,mid) of F
        launch_naive(r, lw[i].Wd1, fdt, rowsBMK, mid, 3, mid, 1, stream);
        if (mid % 16 == 0)
            launch_wmma(fdt, lw[i].Wd2, wpack, Fbuf, rowsBMK, mid, mid, Ctot, 1, stream);
        else
            launch_naive(fdt, lw[i].Wd2, Fbuf, rowsBMK, mid, mid, Ctot, 1, stream);

        // gathered previous features -> columns [mid, Ctot) of F
        if (cf > 0) {
            int total = rowsBMK * cf;
            gather_feats<<<(total + 255) / 256, 256, 0, stream>>>(
                featsPrev, idx, Fbuf, M, K, mid, cf, Ctot, P, total);
        }

        // X = elu(elu(r_flat @ Wx1) @ Wx2) @ Wx3   (rows = B*M, r_flat cols = 3K)
        launch_wmma(r, lw[i].Wx1, wpack, t1, rowsBM, K * K, 3 * K, K * K, 1, stream);
        launch_wmma(t1, lw[i].Wx2, wpack, t2, rowsBM, K * K, K * K, K * K, 1, stream);
        launch_wmma(t2, lw[i].Wx3, wpack, Xm, rowsBM, K * K, K * K, K * K, 0, stream);

        // Fx[b,m] = X(KxK) @ F(KxCtot)
        launch_xf(Xm, Fbuf, Fx, rowsBM, K, Ctot, stream);

        // out_feats = elu(Fx_flat @ Wc)
        launch_wmma(Fx, lw[i].Wc, wpack, fo, rowsBM, co, K * Ctot, co, 1, stream);

        P = M;
        featsPrev = fo;
        ffinal = fo;
    }

    // Final MLP head: h = elu(f @ Wm1); h = elu(h @ Wm2); mean(h @ Wfc + b)
    launch_wmma(ffinal, Wm1, wpack, h1, BN * 128, 256, 384, 256, 1, stream);
    launch_wmma(h1, Wm2, wpack, h2, BN * 128, 128, 256, 128, 1, stream);
    final_fc_mean<<<(BN * 40 + 63) / 64, 64, 0, stream>>>(h2, Wfc, bfc, out, BN * 40);
}